// BatchHighOrderActivation_841813590311
// MI455X (gfx1250) — compile-verified
//
#include <hip/hip_runtime.h>
#include <cstdint>

// Problem constants (from reference)
#define B_TOT   8192
#define I_TOT   1024
#define TI      32          // i-columns per block
#define THREADS 256
#define ROWSTR  132         // padded dwords per i-slice in LDS (128 data + 4 pad)
#define B_PER_BLOCK 64      // 8 warp-rows * 8 iterations

typedef float v4f __attribute__((ext_vector_type(4)));

// compare-exchange on (value, index) pairs, branchless
#define CE(a, b, x, y)                         \
  {                                            \
    bool sw_ = (a) > (b);                      \
    float tf_ = sw_ ? (b) : (a);               \
    (b) = sw_ ? (a) : (b);                     \
    (a) = tf_;                                 \
    int tii_ = sw_ ? (y) : (x);                \
    (y) = sw_ ? (x) : (y);                     \
    (x) = tii_;                                \
  }

__global__ __launch_bounds__(THREADS)
void lovasz_ext_kernel(const float* __restrict__ X,
                       const float* __restrict__ P,
                       float* __restrict__ out)
{
    __shared__ float smem[TI * ROWSTR];   // 16.5 KiB: params[i0..i0+31][16][8], padded rows

    const int tid = threadIdx.x;
    const int i0  = blockIdx.x * TI;
    const int b0  = blockIdx.y * B_PER_BLOCK;

    // ---- Stage params slice (32 i * 512 B = 16 KiB) into LDS via CDNA5
    //      async global->LDS copies (ASYNCcnt path, no VGPR staging). ----
    {
        const char* gbase = (const char*)(P + (size_t)i0 * (16 * 8));
        #pragma unroll
        for (int r = 0; r < 4; ++r) {
            int fofs   = (r * THREADS + tid) * 4;     // dword index in slice [0,4096)
            int i_rel  = fofs >> 7;                   // /128 dwords per i
            int within = fofs & 127;
            uint32_t lds_addr = (uint32_t)(uintptr_t)(&smem[i_rel * ROWSTR + within]);
            uint64_t gaddr    = (uint64_t)(uintptr_t)(gbase + (size_t)fofs * 4);
            asm volatile("global_load_async_to_lds_b128 %0, %1, off"
                         :: "v"(lds_addr), "v"(gaddr) : "memory");
        }
    }

    const int i_lane  = tid & 31;           // lanes span consecutive i -> coalesced X/out
    const int i       = i0 + i_lane;
    const int ldsbase = i_lane * ROWSTR;
    const int brow    = tid >> 5;           // 0..7

    const v4f* Xv = (const v4f*)X;
    v4f*       Ov = (v4f*)out;

    // ---- Issue all 8 X loads (LOADcnt) BEFORE waiting on the async staging
    //      (ASYNCcnt): the two HBM streams overlap instead of serializing. ----
    v4f    xv[8];
    size_t pair[8];
    #pragma unroll
    for (int it = 0; it < 8; ++it) {
        const int b = b0 + it * 8 + brow;
        pair[it] = (size_t)b * I_TOT + i;
        xv[it]   = __builtin_nontemporal_load(Xv + pair[it]);  // 16B/lane, 512B/wave
    }

    asm volatile("s_wait_asynccnt 0x0" ::: "memory");
    __syncthreads();

    #pragma unroll
    for (int it = 0; it < 8; ++it) {
        // ---- sort 4 values, tracking original indices ----
        float v0 = xv[it].x, v1 = xv[it].y, v2 = xv[it].z, v3 = xv[it].w;
        int d0 = 0, d1 = 1, d2 = 2, d3 = 3;
        CE(v0, v1, d0, d1);
        CE(v2, v3, d2, d3);
        CE(v0, v2, d0, d2);
        CE(v1, v3, d1, d3);
        CE(v1, v2, d1, d2);

        // successive differences
        float c0 = v0, c1 = v1 - v0, c2 = v2 - v1, c3 = v3 - v2;

        // suffix sums of (1 << ind): idx[k] = sum_{j>=k} 1<<ind[j]; idx[0]==15 always
        int s1 = 1 << d1, s2 = 1 << d2, s3 = 1 << d3;
        int m3 = s3;
        int m2 = s3 + s2;
        int m1 = m2 + s1;
        // m0 == 15 (permutation of {1,2,4,8} sums to 15)

        // ---- gather 4 rows of 8 from LDS, weighted accumulate ----
        const float* r0 = &smem[ldsbase + 15 * 8];
        const float* r1 = &smem[ldsbase + m1 * 8];
        const float* r2 = &smem[ldsbase + m2 * 8];
        const float* r3 = &smem[ldsbase + m3 * 8];

        v4f p0a = *(const v4f*)(r0);  v4f p0b = *(const v4f*)(r0 + 4);
        v4f p1a = *(const v4f*)(r1);  v4f p1b = *(const v4f*)(r1 + 4);
        v4f p2a = *(const v4f*)(r2);  v4f p2b = *(const v4f*)(r2 + 4);
        v4f p3a = *(const v4f*)(r3);  v4f p3b = *(const v4f*)(r3 + 4);

        v4f acc_lo = c0 * p0a;
        v4f acc_hi = c0 * p0b;
        acc_lo += c1 * p1a;  acc_hi += c1 * p1b;
        acc_lo += c2 * p2a;  acc_hi += c2 * p2b;
        acc_lo += c3 * p3a;  acc_hi += c3 * p3b;

        // 32B per lane, contiguous 1KiB across the wave; stream past caches
        __builtin_nontemporal_store(acc_lo, Ov + pair[it] * 2);
        __builtin_nontemporal_store(acc_hi, Ov + pair[it] * 2 + 1);
    }
}

extern "C" void kernel_launch(void* const* d_in, const int* in_sizes, int n_in,
                              void* d_out, int out_size, void* d_ws, size_t ws_size,
                              hipStream_t stream) {
    const float* X = (const float*)d_in[0];   // (8192, 1024, 4) f32
    const float* P = (const float*)d_in[1];   // (1024, 16, 8) f32
    float* out = (float*)d_out;               // (8192, 1024, 8) f32

    dim3 grid(I_TOT / TI, B_TOT / B_PER_BLOCK);   // (32, 128)
    lovasz_ext_kernel<<<grid, THREADS, 0, stream>>>(X, P, out);
}